// EdgeEncoding_31945966748033
// MI455X (gfx1250) — compile-verified
//
#include <hip/hip_runtime.h>
#include <hip/hip_bf16.h>

typedef float v2f __attribute__((ext_vector_type(2)));
typedef float v4f __attribute__((ext_vector_type(4)));
typedef float v8f __attribute__((ext_vector_type(8)));

// ---------------------------------------------------------------------------
// Zero-fill d_out (64 MB), vectorized float4 stores, grid-stride.
// ---------------------------------------------------------------------------
__global__ void EdgeEncoding_zero_kernel(float* __restrict__ out, long n) {
  const long i      = (long)blockIdx.x * blockDim.x + threadIdx.x;
  const long stride = (long)gridDim.x * blockDim.x;
  const long n4     = n >> 2;
  const float4 z    = make_float4(0.f, 0.f, 0.f, 0.f);
  for (long j = i; j < n4; j += stride) ((float4*)out)[j] = z;
  for (long j = (n4 << 2) + i; j < n; j += stride) out[j] = 0.f;
}

// ---------------------------------------------------------------------------
// Main kernel: one wave32 processes 16 pairs.
//   lanes 0-15  : pair m = lane,     features 0..7
//   lanes 16-31 : pair m = lane-16,  features 8..15
// Per-lane masked partial dot accumulated in VALU; the lane/lane+16 halves are
// combined with a single V_WMMA_F32_16X16X4_F32 (A = partial sums in the
// documented 16x4 layout, B = ones so B's K-striping is irrelevant).
// D[m, n] (same for every n) = full dot-sum for pair m.
// ---------------------------------------------------------------------------
__global__ __launch_bounds__(256) void EdgeEncoding_main_kernel(
    const float* __restrict__ edge_attr,     // [E, 16]
    const float* __restrict__ edge_weights,  // [5, 16]
    const int*   __restrict__ path_idx,      // [P, 5]
    const int*   __restrict__ path_lens,     // [P]
    const int*   __restrict__ pair_id,       // [P]
    float*       __restrict__ out,           // [N, N]
    long P, int N)
{
  const int  lane = threadIdx.x & 31;
  const long wid  = ((long)blockIdx.x * blockDim.x + threadIdx.x) >> 5;
  const long base = wid * 16;

  const int  m  = lane & 15;
  long p        = base + m;
  const bool pv = (p < P);
  const long pc = pv ? p : (P - 1);          // clamp: safe loads, masked result
  const int  len = pv ? path_lens[pc] : 0;

  const int fo = (lane >> 4) * 8;            // feature offset: 0 or 8

  float acc = 0.f;
#pragma unroll
  for (int l = 0; l < 5; ++l) {
    const int e = path_idx[pc * 5 + l];      // edge gather index (L2-resident table)
    const float* gp = edge_attr    + (long)e * 16 + fo;
    const float* wp = edge_weights + l * 16 + fo;
    const v4f g0 = *(const v4f*)(gp);
    const v4f g1 = *(const v4f*)(gp + 4);
    const v4f w0 = *(const v4f*)(wp);
    const v4f w1 = *(const v4f*)(wp + 4);
    float d = g0.x * w0.x + g0.y * w0.y + g0.z * w0.z + g0.w * w0.w
            + g1.x * w1.x + g1.y * w1.y + g1.z * w1.z + g1.w * w1.w;
    acc += (l < len) ? d : 0.f;              // ragged-length mask (select, no divergence)
  }

  // Cross-half reduction via WMMA: D = A(16x4) * ones(4x16) + 0.
  // A layout (ISA 7.12.2): lane m -> A[m,0],A[m,1]; lane m+16 -> A[m,2],A[m,3].
  v2f a;  a.x = acc; a.y = 0.f;
  v2f b1; b1.x = 1.f; b1.y = 1.f;
  v8f c = {0.f, 0.f, 0.f, 0.f, 0.f, 0.f, 0.f, 0.f};
  c = __builtin_amdgcn_wmma_f32_16x16x4_f32(
        /*neg_a=*/false, a, /*neg_b=*/false, b1,
        /*c_mod=*/(short)0, c, /*reuse_a=*/false, /*reuse_b=*/false);

  // C/D layout: lane n (0-15) VGPR r = D[r,n]; lane 16+n VGPR r = D[8+r,n].
  // Pair j (0-7)  -> lane j,    c[j]   (column n=j)
  // Pair 8+q      -> lane 16+q, c[q]   (column n=q)
  const int t = lane & 7;
  float v = c[0];
  v = (t == 1) ? c[1] : v;
  v = (t == 2) ? c[2] : v;
  v = (t == 3) ? c[3] : v;
  v = (t == 4) ? c[4] : v;
  v = (t == 5) ? c[5] : v;
  v = (t == 6) ? c[6] : v;
  v = (t == 7) ? c[7] : v;

  const bool storer = (lane < 8) || (lane >= 16 && lane < 24);
  const int  pm     = (lane < 8) ? lane : (lane - 8);   // pair slot 0..15
  const long sp     = base + pm;
  if (storer && sp < P) {
    const int   len2 = path_lens[sp];
    const float mean = (len2 > 0) ? (v / (float)len2) : 0.f;
    const int   pid  = pair_id[sp];
    const int   src  = pid % N;
    const int   dst  = pid / N;
    out[(long)src * N + dst] = mean;         // scatter; output fits in 192MB L2
  }
}

extern "C" void kernel_launch(void* const* d_in, const int* in_sizes, int n_in,
                              void* d_out, int out_size, void* d_ws, size_t ws_size,
                              hipStream_t stream) {
  // setup_inputs() order: x, edge_attr, edge_weights, path_idx, path_lens, pair_id
  const float* edge_attr    = (const float*)d_in[1];
  const float* edge_weights = (const float*)d_in[2];
  const int*   path_idx     = (const int*)d_in[3];
  const int*   path_lens    = (const int*)d_in[4];
  const int*   pair_id      = (const int*)d_in[5];
  float* out = (float*)d_out;

  const long P = (long)in_sizes[4];          // number of pairs
  const int  N = in_sizes[0] / 16;           // x is [N, EDGE_DIM=16]

  // 1) zero the [N,N] output (d_out is poisoned by the harness)
  EdgeEncoding_zero_kernel<<<8192, 256, 0, stream>>>(out, (long)out_size);

  // 2) gather / dot / mean / scatter; one wave per 16 pairs
  const long waves  = (P + 15) / 16;
  const long blocks = (waves + 7) / 8;       // 8 wave32 per 256-thread block
  EdgeEncoding_main_kernel<<<(int)blocks, 256, 0, stream>>>(
      edge_attr, edge_weights, path_idx, path_lens, pair_id, out, P, N);
}